// Transformer_71279277244771
// MI455X (gfx1250) — compile-verified
//
#include <hip/hip_runtime.h>
#include <hip/hip_bf16.h>
#include <stdint.h>

// ---------------------------------------------------------------------------
// RetinaNet post-processing for MI455X (gfx1250, wave32).
//   K1 decode     : elementwise anchor regression + clip           (vector)
//   K2 topk_select: per-class 2-level 11-bit radix select of top-1000
//                   (8-bank LDS f32 histograms, reduced with
//                    V_WMMA_F32_16X16X4_F32, A = ones -> exact f32 sums)
//   K3 nms        : block-per-class greedy NMS fully in LDS; selected
//                   scores staged LDS-side via the Tensor Data Mover
//                   (tensor_load_to_lds + s_wait_tensorcnt).
// ---------------------------------------------------------------------------

#define NANCH   196416      // 9*(128^2+64^2+32^2+16^2+8^2)
#define NCLS    80
#define KSEL    1024        // padded PRE_NMS working set (top-1000 + slack)
#define TOPK_F  1000.0f
#define MAXDET  100
#define NEGF    (-1.0e9f)

typedef __attribute__((ext_vector_type(2)))  float    v2f;
typedef __attribute__((ext_vector_type(8)))  float    v8f;
typedef __attribute__((ext_vector_type(4)))  unsigned u32x4;
typedef __attribute__((ext_vector_type(8)))  int      i32x8;
typedef __attribute__((ext_vector_type(4)))  int      i32x4;

// ---------------------------------------------------------------- K1: decode
__global__ __launch_bounds__(256)
void decode_kernel(const float4* __restrict__ anchors,
                   const float*  __restrict__ reg,
                   float4*       __restrict__ out)
{
    int n = blockIdx.x * blockDim.x + threadIdx.x;
    if (n >= NANCH) return;
    float4 a = anchors[n];
    float w  = a.z - a.x, h = a.w - a.y;
    float cx = a.x + 0.5f * w, cy = a.y + 0.5f * h;
    float dx = reg[0 * NANCH + n] * 0.1f;
    float dy = reg[1 * NANCH + n] * 0.1f;
    float dw = reg[2 * NANCH + n] * 0.2f;
    float dh = reg[3 * NANCH + n] * 0.2f;
    float pcx = cx + dx * w, pcy = cy + dy * h;
    float pw  = __expf(dw) * w, ph = __expf(dh) * h;
    float x1 = fmaxf(pcx - 0.5f * pw, 0.0f);
    float y1 = fmaxf(pcy - 0.5f * ph, 0.0f);
    float x2 = fminf(pcx + 0.5f * pw, 1024.0f);
    float y2 = fminf(pcy + 0.5f * ph, 1024.0f);
    out[n] = make_float4(x1, y1, x2, y2);
}

// ------------------------------------------------ WMMA 8-bank hist reduction
// Sums 8 per-bank f32 histograms (2048 bins, each shared by 4 waves) into outR using
// V_WMMA_F32_16X16X4_F32 with A = all-ones: D[m][n] = sum_k B[k][n].
// Counts < 2^24 so f32 accumulation is exact. Executed by one full wave
// (EXEC all-ones, as WMMA requires).
__device__ inline void wmma_reduce_hist(float (*hist)[2048], float* outR)
{
    const int  lane = threadIdx.x & 31;
    const int  nsel = lane & 15;
    const bool hi   = lane >= 16;
    v2f a; a[0] = 1.0f; a[1] = 1.0f;                 // 16x4 ones
    for (int g = 0; g < 128; ++g) {
        int col = g * 16 + nsel;
        v2f b0, b1;
        b0[0] = hi ? hist[2][col] : hist[0][col];
        b0[1] = hi ? hist[3][col] : hist[1][col];
        b1[0] = hi ? hist[6][col] : hist[4][col];
        b1[1] = hi ? hist[7][col] : hist[5][col];
        v8f acc = {};
        acc = __builtin_amdgcn_wmma_f32_16x16x4_f32(false, a, false, b0,
                                                    (short)0, acc, false, false);
        acc = __builtin_amdgcn_wmma_f32_16x16x4_f32(false, a, false, b1,
                                                    (short)0, acc, false, false);
        if (!hi) outR[col] = acc[0];                 // D row 0, N = lane
    }
}

// ------------------------------------------------------- K2: top-k selection
__global__ __launch_bounds__(1024)
void topk_select(const float*  __restrict__ cls,
                 const float4* __restrict__ boxes,
                 float*        __restrict__ sel_s,
                 float4*       __restrict__ sel_b)
{
    const int c = blockIdx.x, t = threadIdx.x;
    __shared__ float    hist[8][2048];   // 8 banks, 64 KB
    __shared__ float    histR[2048];
    __shared__ int      s_p1;
    __shared__ float    s_cumAbove;
    __shared__ unsigned s_pivotKey;
    __shared__ int      s_cnt;

    const float* sc   = cls + (size_t)c * NANCH;
    const int    bank = (t >> 5) & 7;
    float*       hf   = &hist[0][0];

    for (int i = t; i < 8 * 2048; i += 1024) hf[i] = 0.0f;
    if (t == 0) { s_cnt = 0; s_pivotKey = 0u; s_p1 = -1; }
    __syncthreads();

    // Pass A: coarse histogram on key bits [31:21] (all candidates positive).
    for (int n = t; n < NANCH; n += 1024) {
        float s = sc[n];
        if (s > 0.05f)
            atomicAdd(&hist[bank][__float_as_uint(s) >> 21], 1.0f);
    }
    __syncthreads();
    if (t < 32) wmma_reduce_hist(hist, histR);
    __syncthreads();
    if (t == 0) {
        float cum = 0.0f;
        for (int b = 2047; b >= 0; --b) {
            float hb = histR[b];
            if (cum + hb >= TOPK_F) { s_p1 = b; s_cumAbove = cum; break; }
            cum += hb;
        }
    }
    __syncthreads();
    const int p1 = s_p1;

    // Pass B: refine inside pivot bucket on key bits [20:10].
    for (int i = t; i < 8 * 2048; i += 1024) hf[i] = 0.0f;
    __syncthreads();
    if (p1 >= 0) {
        for (int n = t; n < NANCH; n += 1024) {
            float s = sc[n];
            if (s > 0.05f) {
                unsigned k = __float_as_uint(s);
                if ((int)(k >> 21) == p1)
                    atomicAdd(&hist[bank][(k >> 10) & 2047], 1.0f);
            }
        }
    }
    __syncthreads();
    if (t < 32) wmma_reduce_hist(hist, histR);
    __syncthreads();
    if (t == 0 && p1 >= 0) {
        float    cum = s_cumAbove;
        unsigned pk  = (unsigned)p1 << 11;
        for (int b = 2047; b >= 0; --b) {
            cum += histR[b];
            if (cum >= TOPK_F) { pk = ((unsigned)p1 << 11) | (unsigned)b; break; }
        }
        s_pivotKey = pk;
    }
    __syncthreads();
    const unsigned pivotKey = s_pivotKey;

    // Init output slots (NEG padding), then compact survivors.
    float*  ss = sel_s + (size_t)c * KSEL;
    float4* sb = sel_b + (size_t)c * KSEL;
    ss[t] = NEGF;
    sb[t] = make_float4(0.0f, 0.0f, 0.0f, 0.0f);
    __syncthreads();

    for (int n = t; n < NANCH; n += 1024) {
        float s = sc[n];
        if (s > 0.05f && (__float_as_uint(s) >> 10) >= pivotKey) {
            int slot = atomicAdd(&s_cnt, 1);
            if (slot < KSEL) { ss[slot] = s; sb[slot] = boxes[n]; }
        }
    }
}

// --------------------------------------------------------------- K3: NMS
struct SMemN {                      // single shared object -> LDS offset 0
    float sc[KSEL];                 // scores (TDM target at LDS addr 0)
    float bx1[KSEL], by1[KSEL], bx2[KSEL], by2[KSEL], ar[KSEL];
    float rv[KSEL];
    int   ri[KSEL];
    float sel_v[MAXDET];
    float sel_x1[MAXDET], sel_y1[MAXDET], sel_x2[MAXDET], sel_y2[MAXDET];
    int   best;
};

__global__ __launch_bounds__(1024)
void nms_kernel(const float*  __restrict__ sel_s,
                const float4* __restrict__ sel_b,
                float*        __restrict__ out)
{
    const int c = blockIdx.x, t = threadIdx.x;
    __shared__ SMemN sm;

    // --- TDM: async 1-D tile load of the 1024 selected scores into LDS.
    //     1 descriptor, data_size=4B, tile_dim0=tensor_dim0=stride=1024,
    //     lds_addr=0 (sm.sc). Issued by wave 0 only; TENSORcnt drained
    //     before the block barrier publishes LDS to all waves.
    //     6-arg builtin on this toolchain: groups 0..4 + cpol; groups 2..4
    //     are unused for a 1-D tensor and passed zero-filled.
    if (t < 32) {
        uint64_t ga = (uint64_t)(const void*)(sel_s + (size_t)c * KSEL);
        u32x4 g0;
        g0[0] = 1u;                                        // count=1, user mode
        g0[1] = 0u;                                        // lds_addr = 0
        g0[2] = (unsigned)(ga & 0xFFFFFFFFu);              // global_addr lo
        g0[3] = (unsigned)((ga >> 32) & 0x1FFFFFFu)        // global_addr hi
              | (2u << 30);                                // type = 2 (image)
        i32x8 g1;
        g1[0] = (int)(2u << 16);          // wg_mask=0, data_size=2 (4 bytes)
        g1[1] = (int)((KSEL & 0xFFFFu) << 16);   // tensor_dim0 lo16
        g1[2] = (int)(1u << 16);          // tensor_dim0 hi=0, tensor_dim1=1
        g1[3] = (int)((KSEL & 0xFFFFu) << 16);   // tile_dim0 = 1024
        g1[4] = 1;                        // tile_dim1 = 1, tile_dim2 = 0
        g1[5] = KSEL;                     // tensor_dim0_stride lo32
        g1[6] = 0;
        g1[7] = 0;
        i32x4 g2 = {0, 0, 0, 0};
        i32x4 g3 = {0, 0, 0, 0};
        i32x8 g4 = {0, 0, 0, 0, 0, 0, 0, 0};
        __builtin_amdgcn_tensor_load_to_lds(g0, g1, g2, g3, g4, 0);
        __builtin_amdgcn_s_wait_tensorcnt(0);
    }

    // Boxes via regular vector loads; precompute areas.
    float4 b = sel_b[(size_t)c * KSEL + t];
    sm.bx1[t] = b.x; sm.by1[t] = b.y; sm.bx2[t] = b.z; sm.by2[t] = b.w;
    sm.ar[t]  = (b.z - b.x) * (b.w - b.y);
    __syncthreads();

    // Greedy NMS: 100 serial (argmax, suppress) steps, all in LDS.
    for (int it = 0; it < MAXDET; ++it) {
        sm.rv[t] = sm.sc[t];
        sm.ri[t] = t;
        __syncthreads();
        for (int off = 512; off > 0; off >>= 1) {
            if (t < off) {
                float vo = sm.rv[t + off]; int io = sm.ri[t + off];
                float v  = sm.rv[t];       int i  = sm.ri[t];
                if (vo > v || (vo == v && io < i)) { sm.rv[t] = vo; sm.ri[t] = io; }
            }
            __syncthreads();
        }
        if (t == 0) {
            int bi = sm.ri[0];
            sm.best       = bi;
            sm.sel_v[it]  = sm.rv[0];
            sm.sel_x1[it] = sm.bx1[bi]; sm.sel_y1[it] = sm.by1[bi];
            sm.sel_x2[it] = sm.bx2[bi]; sm.sel_y2[it] = sm.by2[bi];
            sm.sc[bi] = NEGF;
        }
        __syncthreads();
        int   bi  = sm.best;
        float ix1 = fmaxf(sm.bx1[bi], sm.bx1[t]);
        float iy1 = fmaxf(sm.by1[bi], sm.by1[t]);
        float ix2 = fminf(sm.bx2[bi], sm.bx2[t]);
        float iy2 = fminf(sm.by2[bi], sm.by2[t]);
        float inter = fmaxf(ix2 - ix1, 0.0f) * fmaxf(iy2 - iy1, 0.0f);
        float iou   = inter / (sm.ar[bi] + sm.ar[t] - inter + 1e-8f);
        if (iou > 0.5f) sm.sc[t] = NEGF;
        __syncthreads();
    }

    // Emit: scores[8000] | cls[8000] | boxes[8000][4], class-major.
    if (t < MAXDET) {
        float v     = sm.sel_v[t];
        bool  valid = v > 0.5f * NEGF;
        int   o     = c * MAXDET + t;
        out[o]                       = valid ? v        : 0.0f;
        out[NCLS * MAXDET + o]       = valid ? (float)c : -1.0f;
        float* bo = out + 2 * NCLS * MAXDET + o * 4;
        bo[0] = valid ? sm.sel_x1[t] : 0.0f;
        bo[1] = valid ? sm.sel_y1[t] : 0.0f;
        bo[2] = valid ? sm.sel_x2[t] : 0.0f;
        bo[3] = valid ? sm.sel_y2[t] : 0.0f;
    }
}

// --------------------------------------------------------------------- host
extern "C" void kernel_launch(void* const* d_in, const int* in_sizes, int n_in,
                              void* d_out, int out_size, void* d_ws, size_t ws_size,
                              hipStream_t stream)
{
    // setup_inputs order: image, cls_pred, reg_pred, anchors
    const float*  cls_pred = (const float*)d_in[1];
    const float*  reg_pred = (const float*)d_in[2];
    const float4* anchors  = (const float4*)d_in[3];
    float*        out      = (float*)d_out;

    // Workspace layout (~4.8 MB)
    float4* boxes_dec = (float4*)d_ws;                         // N * 16 B
    float*  sel_s     = (float*)(boxes_dec + NANCH);           // C*KSEL * 4 B
    float4* sel_b     = (float4*)(sel_s + NCLS * KSEL);        // C*KSEL * 16 B

    decode_kernel<<<(NANCH + 255) / 256, 256, 0, stream>>>(anchors, reg_pred,
                                                           boxes_dec);
    topk_select<<<NCLS, 1024, 0, stream>>>(cls_pred, boxes_dec, sel_s, sel_b);
    nms_kernel<<<NCLS, 1024, 0, stream>>>(sel_s, (const float4*)sel_b, out);
}